// ChannelAttention_1726576854491
// MI455X (gfx1250) — compile-verified
//
#include <hip/hip_runtime.h>
#include <hip/hip_bf16.h>

typedef __attribute__((ext_vector_type(2))) float v2f;
typedef __attribute__((ext_vector_type(8))) float v8f;
typedef __attribute__((ext_vector_type(4))) unsigned int v4u;
typedef __attribute__((ext_vector_type(4))) int v4i;
typedef __attribute__((ext_vector_type(8))) int v8i;

// D = A(16x4) * B(4x16) + C, fp32, wave32 WMMA (CDNA5)
__device__ __forceinline__ v8f wmma4(v2f a, v2f b, v8f c) {
    return __builtin_amdgcn_wmma_f32_16x16x4_f32(
        /*neg_a=*/false, a, /*neg_b=*/false, b,
        /*c_mod=*/(short)0, c, /*reuse_a=*/false, /*reuse_b=*/false);
}

// ---------------------------------------------------------------------------
// Tensor Data Mover: 2D tile (tile_y rows x tile_x elems, fp32) from global
// row-major [.., row_stride_elems] into contiguous LDS at lds_off.
// D# packing per CDNA5 ISA ch.8 (group0: count/lds/global/type; group1:
// data_size, tensor dims, tile dims, dim0 stride). tensor_dim1 set huge so
// no OOB clipping occurs (our tiles are always in-bounds).
// ---------------------------------------------------------------------------
__device__ __forceinline__ void tdm_load_tile_2d(unsigned lds_off, const void* gptr,
                                                 unsigned tile_x, unsigned tile_y,
                                                 unsigned row_stride_elems,
                                                 unsigned tensor_dim0) {
    unsigned long long ga = (unsigned long long)(uintptr_t)gptr;
    v4u g0;
    g0.x = 1u;                                                  // count=1, user D#
    g0.y = lds_off;                                             // lds_addr (bytes)
    g0.z = (unsigned)ga;                                        // global_addr[31:0]
    g0.w = ((unsigned)(ga >> 32) & 0x01FFFFFFu) | 0x80000000u;  // ga[56:32] | type=2
    const unsigned td1 = 1u << 20;                              // tensor_dim1: large
    unsigned long long s0 = row_stride_elems;                   // dim0 stride (elems)
    v8i g1;
    g1[0] = (int)(2u << 16);                                    // data_size = 4 bytes
    g1[1] = (int)((tensor_dim0 & 0xFFFFu) << 16);               // tensor_dim0[15:0]
    g1[2] = (int)((tensor_dim0 >> 16) | ((td1 & 0xFFFFu) << 16));
    g1[3] = (int)((td1 >> 16) | (tile_x << 16));                // tile_dim0
    g1[4] = (int)(tile_y & 0xFFFFu);                            // tile_dim1 (dim2=0)
    g1[5] = (int)(unsigned)s0;                                  // dim0_stride[31:0]
    g1[6] = (int)((s0 >> 32) & 0xFFFFu);                        // dim0_stride[47:32]
    g1[7] = 0;                                                  // dim1_stride unused
    v4i z4 = {};
#if __clang_major__ >= 23
    v8i z8 = {};
    __builtin_amdgcn_tensor_load_to_lds(g0, g1, z4, z4, z8, 0);
#else
    __builtin_amdgcn_tensor_load_to_lds(g0, g1, z4, z4, 0);
#endif
}

// ---------------------------------------------------------------------------
// Tiled fp32 WMMA GEMM: C[M,Nw] = A[M,K] * Bw[K,Nw] (+ bias)
// Block tile 128x64, BK=32, 256 threads = 8 waves, each wave 16 rows x 64 cols.
// Tiles staged by the Tensor Data Mover, double-buffered: wave 0 posts the
// descriptors for tile i+1, waits TENSORcnt<=2 (tile i landed, i+1 in flight),
// barrier publishes LDS, all waves issue WMMAs while DMA fills the other buffer.
// ---------------------------------------------------------------------------
template <bool WITH_BIAS>
__global__ __launch_bounds__(256) void gemm_tiled(
    const float* __restrict__ A, const float* __restrict__ Bw,
    const float* __restrict__ bias, float* __restrict__ C,
    int K, int Nw) {
    __shared__ float As[2][128 * 32];
    __shared__ float Bs[2][32 * 64];
    const int tid  = threadIdx.x;
    const int lane = tid & 31;
    const int wave = tid >> 5;
    const int m    = lane & 15;
    const int hi   = lane >> 4;

    const size_t row0 = (size_t)blockIdx.y * 128;
    const int    col0 = blockIdx.x * 64;
    const int    nk   = K >> 5;

    v8f acc[4];
#pragma unroll
    for (int t = 0; t < 4; ++t) acc[t] = {};

    if (wave == 0) {
        tdm_load_tile_2d((unsigned)(uintptr_t)&As[0][0], A + row0 * (size_t)K,
                         32, 128, (unsigned)K, (unsigned)K);
        tdm_load_tile_2d((unsigned)(uintptr_t)&Bs[0][0], Bw + col0,
                         64, 32, (unsigned)Nw, (unsigned)Nw);
    }

    for (int i = 0; i < nk; ++i) {
        if (wave == 0) {
            if (i + 1 < nk) {
                const int k1 = (i + 1) * 32;
                const int b1 = (i + 1) & 1;
                tdm_load_tile_2d((unsigned)(uintptr_t)&As[b1][0],
                                 A + row0 * (size_t)K + k1, 32, 128,
                                 (unsigned)K, (unsigned)K);
                tdm_load_tile_2d((unsigned)(uintptr_t)&Bs[b1][0],
                                 Bw + (size_t)k1 * Nw + col0, 64, 32,
                                 (unsigned)Nw, (unsigned)Nw);
                __builtin_amdgcn_s_wait_tensorcnt(2);   // tile i complete
            } else {
                __builtin_amdgcn_s_wait_tensorcnt(0);   // last tile complete
            }
        }
        __syncthreads();
        const float* as = &As[i & 1][0];
        const float* bs = &Bs[i & 1][0];
#pragma unroll
        for (int kk = 0; kk < 32; kk += 4) {
            // A-frag: a[j] = A[wave*16 + m][kk + j + 2*hi]
            v2f av = *(const v2f*)&as[(wave * 16 + m) * 32 + kk + 2 * hi];
#pragma unroll
            for (int t = 0; t < 4; ++t) {
                // B-frag: b[j] = B[kk + j + 2*hi][t*16 + m]
                v2f bv;
                bv.x = bs[(kk + 2 * hi) * 64 + t * 16 + m];
                bv.y = bs[(kk + 1 + 2 * hi) * 64 + t * 16 + m];
                acc[t] = wmma4(av, bv, acc[t]);
            }
        }
        __syncthreads();
    }

    // Store: D vgpr r, lane -> row m' = r + 8*hi, col n = t*16 + m
#pragma unroll
    for (int t = 0; t < 4; ++t) {
#pragma unroll
        for (int r = 0; r < 8; ++r) {
            size_t row = row0 + wave * 16 + r + 8 * hi;
            int    col = col0 + t * 16 + m;
            float  v   = acc[t][r];
            if (WITH_BIAS) v += bias[col];
            C[row * (size_t)Nw + col] = v;
        }
    }
}

// ---------------------------------------------------------------------------
// Partial sum-of-squares over token axis for q,k channels (cols 0..1535 of qkv)
// part[z][b*1536 + col], z = 8 slices of 512 tokens. Deterministic (no atomics).
// ---------------------------------------------------------------------------
__global__ void sumsq_part_kernel(const float* __restrict__ qkv,
                                  float* __restrict__ part) {
    const int b = blockIdx.x, cg = blockIdx.y, z = blockIdx.z;
    const int col = cg * 128 + threadIdx.x;  // 0..1535
    const float* base = qkv + ((size_t)b * 4096 + (size_t)z * 512) * 2304 + col;
    float s = 0.f;
    for (int n = 0; n < 512; ++n) {
        float x = base[(size_t)n * 2304];
        s += x * x;
    }
    part[(size_t)z * 24576 + (size_t)b * 1536 + col] = s;
}

__global__ void rn_kernel(const float* __restrict__ part, float* __restrict__ rn) {
    int i = blockIdx.x * 256 + threadIdx.x;  // < 24576
    float s = 0.f;
#pragma unroll
    for (int z = 0; z < 8; ++z) s += part[(size_t)z * 24576 + i];
    float nrm = sqrtf(s);
    rn[i] = 1.f / fmaxf(nrm, 1e-12f);
}

// ---------------------------------------------------------------------------
// Partial Gram: G[dd,ee] = sum_n q[dd,n]*k[ee,n] over a 1024-token slice.
// 192 threads = 6 waves; wave w owns dd rows [16w,16w+16) -> 6 WMMA col tiles.
// ---------------------------------------------------------------------------
__global__ __launch_bounds__(192) void attn_qk_kernel(
    const float* __restrict__ qkv, float* __restrict__ Gpart) {
    const int h = blockIdx.x, b = blockIdx.y, z = blockIdx.z;  // z in 0..3
    __shared__ float qs[96 * 64];
    __shared__ float ks[96 * 64];
    const int tid = threadIdx.x, lane = tid & 31, wave = tid >> 5;
    const int m = lane & 15, hi = lane >> 4;

    v8f acc[6];
#pragma unroll
    for (int t = 0; t < 6; ++t) acc[t] = {};

    const size_t rowbase = (size_t)b * 4096 + (size_t)z * 1024;
    const int dd = tid % 96, half = tid / 96;

    for (int nc = 0; nc < 16; ++nc) {
        const float* src = qkv + (rowbase + nc * 64) * 2304 + h * 96 + dd;
        for (int nn = half; nn < 64; nn += 2) {
            qs[dd * 64 + nn] = src[(size_t)nn * 2304];
            ks[dd * 64 + nn] = src[(size_t)nn * 2304 + 768];
        }
        __syncthreads();
#pragma unroll
        for (int kk = 0; kk < 64; kk += 4) {
            v2f av = *(const v2f*)&qs[(wave * 16 + m) * 64 + kk + 2 * hi];
#pragma unroll
            for (int t = 0; t < 6; ++t) {
                v2f bv = *(const v2f*)&ks[(t * 16 + m) * 64 + kk + 2 * hi];
                acc[t] = wmma4(av, bv, acc[t]);
            }
        }
        __syncthreads();
    }

    float* g = Gpart + ((size_t)z * 128 + (size_t)b * 8 + h) * 9216;
#pragma unroll
    for (int t = 0; t < 6; ++t) {
#pragma unroll
        for (int r = 0; r < 8; ++r) {
            int ddr = wave * 16 + r + 8 * hi;
            int ee  = t * 16 + m;
            g[ddr * 96 + ee] = acc[t][r];
        }
    }
}

// ---------------------------------------------------------------------------
// Sum 4 partials, scale by rnq*rnk*temperature, row-softmax over ee -> P
// ---------------------------------------------------------------------------
__global__ __launch_bounds__(96) void softmax_kernel(
    const float* __restrict__ Gpart, const float* __restrict__ rn,
    const float* __restrict__ temperature, float* __restrict__ P) {
    const int h = blockIdx.x, b = blockIdx.y;
    __shared__ float S[96 * 96];
    const int dd = threadIdx.x;  // 0..95
    const float rq  = rn[b * 1536 + h * 96 + dd];
    const float tmp = temperature[h];
    const size_t go = ((size_t)b * 8 + h) * 9216;

    float mx = -3.4e38f;
    for (int ee = 0; ee < 96; ++ee) {
        float gsum = 0.f;
#pragma unroll
        for (int zz = 0; zz < 4; ++zz)
            gsum += Gpart[(size_t)zz * 128 * 9216 + go + dd * 96 + ee];
        float rk = rn[b * 1536 + 768 + h * 96 + ee];
        float s  = gsum * rq * rk * tmp;
        S[dd * 96 + ee] = s;
        mx = fmaxf(mx, s);
    }
    float sum = 0.f;
    for (int ee = 0; ee < 96; ++ee) {
        float e = __expf(S[dd * 96 + ee] - mx);
        S[dd * 96 + ee] = e;
        sum += e;
    }
    float inv = 1.f / sum;
    for (int ee = 0; ee < 96; ++ee)
        P[go + dd * 96 + ee] = S[dd * 96 + ee] * inv;
}

// ---------------------------------------------------------------------------
// out[b, n, h*96+dd] = sum_ee P[dd,ee] * v[ee,n]; 512-token slice per WG.
// ---------------------------------------------------------------------------
__global__ __launch_bounds__(192) void attn_pv_kernel(
    const float* __restrict__ qkv, const float* __restrict__ P,
    float* __restrict__ out) {
    const int h = blockIdx.x, b = blockIdx.y, z = blockIdx.z;  // z 0..7
    __shared__ float Ps[96 * 96];
    __shared__ float vs[96 * 64];
    const int tid = threadIdx.x, lane = tid & 31, wave = tid >> 5;
    const int m = lane & 15, hi = lane >> 4;

    const float* psrc = P + ((size_t)b * 8 + h) * 9216;
    for (int i = tid; i < 9216; i += 192) Ps[i] = psrc[i];

    const size_t rowbase = (size_t)b * 4096 + (size_t)z * 512;
    const int dd = tid % 96, half = tid / 96;

    for (int nc = 0; nc < 8; ++nc) {
        __syncthreads();
        const float* src = qkv + (rowbase + nc * 64) * 2304 + 1536 + h * 96 + dd;
        for (int nn = half; nn < 64; nn += 2)
            vs[dd * 64 + nn] = src[(size_t)nn * 2304];
        __syncthreads();

        v8f acc[4];
#pragma unroll
        for (int t = 0; t < 4; ++t) acc[t] = {};
#pragma unroll
        for (int kk = 0; kk < 96; kk += 4) {
            v2f av = *(const v2f*)&Ps[(wave * 16 + m) * 96 + kk + 2 * hi];
#pragma unroll
            for (int t = 0; t < 4; ++t) {
                v2f bv;
                bv.x = vs[(kk + 2 * hi) * 64 + t * 16 + m];
                bv.y = vs[(kk + 1 + 2 * hi) * 64 + t * 16 + m];
                acc[t] = wmma4(av, bv, acc[t]);
            }
        }
#pragma unroll
        for (int t = 0; t < 4; ++t) {
#pragma unroll
            for (int r = 0; r < 8; ++r) {
                int ddr  = wave * 16 + r + 8 * hi;
                int ncol = nc * 64 + t * 16 + m;
                out[(rowbase + ncol) * 768 + h * 96 + ddr] = acc[t][r];
            }
        }
    }
}

// ---------------------------------------------------------------------------
extern "C" void kernel_launch(void* const* d_in, const int* in_sizes, int n_in,
                              void* d_out, int out_size, void* d_ws, size_t ws_size,
                              hipStream_t stream) {
    const float* x      = (const float*)d_in[0];  // [16,4096,768]
    const float* w_qkv  = (const float*)d_in[1];  // [768,2304]
    const float* temp   = (const float*)d_in[2];  // [8,1,1]
    const float* w_proj = (const float*)d_in[3];  // [768,768]
    const float* b_proj = (const float*)d_in[4];  // [768]
    float* out = (float*)d_out;                   // [16,4096,768]

    char* ws = (char*)d_ws;
    float* qkv      = (float*)ws;                        // 603,979,776 B
    float* attn_out = (float*)(ws + 603979776ULL);       // 201,326,592 B
    float* part     = (float*)(ws + 805306368ULL);       //     786,432 B
    float* rn       = (float*)(ws + 806092800ULL);       //      98,304 B
    float* Gpart    = (float*)(ws + 806191104ULL);       //  18,874,368 B
    float* P        = (float*)(ws + 825065472ULL);       //   4,718,592 B

    // 1) qkv = x @ w_qkv      [65536,768] x [768,2304]
    gemm_tiled<false><<<dim3(36, 512), 256, 0, stream>>>(x, w_qkv, nullptr, qkv, 768, 2304);
    // 2) channel L2 norms of q,k (two-stage, deterministic)
    sumsq_part_kernel<<<dim3(16, 12, 8), 128, 0, stream>>>(qkv, part);
    rn_kernel<<<dim3(96), 256, 0, stream>>>(part, rn);
    // 3) Gram partials G = q k^T over token slices
    attn_qk_kernel<<<dim3(8, 16, 4), 192, 0, stream>>>(qkv, Gpart);
    // 4) scale + softmax -> P
    softmax_kernel<<<dim3(8, 16), 96, 0, stream>>>(Gpart, rn, temp, P);
    // 5) out_attn = P @ v, written in [B,N,C] layout
    attn_pv_kernel<<<dim3(8, 16, 8), 192, 0, stream>>>(qkv, P, attn_out);
    // 6) final projection + bias
    gemm_tiled<true><<<dim3(12, 512), 256, 0, stream>>>(attn_out, w_proj, b_proj, out, 768, 768);
    (void)in_sizes; (void)n_in; (void)out_size; (void)ws_size;
}